// EmbeddingShard_77318001262953
// MI455X (gfx1250) — compile-verified
//
#include <hip/hip_runtime.h>
#include <hip/hip_bf16.h>
#include <stdint.h>
#include <stddef.h>

// out[t, :] = 0.125f * W_full[tok[t], :] + 0.125f * sum_s b[s, :]
// Pure HBM-bandwidth gather: ~512 MB of traffic -> ~22 us roofline @ 23.3 TB/s.

typedef float f32x4 __attribute__((ext_vector_type(4)));

#define EMB_D_MODEL 4096
#define EMB_SHARDS  8
#define EMB_THREADS 256
#define V4_ROW (EMB_D_MODEL / 4)      // 1024 float4 per row
#define V4_THR (V4_ROW / EMB_THREADS) // 4 float4 per thread

#if defined(__gfx1250__) && __has_builtin(__builtin_amdgcn_global_load_async_to_lds_b128)
#define HAVE_ASYNC_LDS 1
// Builtin signature is (v4i AS1*, v4i AS3*, imm int, imm int); v4i is the
// GCC-style vector_size(16) int (matches the hipcc diagnostic exactly).
typedef int i32x4 __attribute__((vector_size(16)));
typedef __attribute__((address_space(1))) i32x4 gbl_i32x4;
typedef __attribute__((address_space(3))) i32x4 lds_i32x4;
#else
#define HAVE_ASYNC_LDS 0
#endif

// Prologue: bias_pre[d] = 0.125 * sum_s b[s][d]   (4096 floats -> d_ws)
__global__ __launch_bounds__(EMB_THREADS) void bias_sum_kernel(
    const float* __restrict__ b, float* __restrict__ bias_pre) {
  const int d = blockIdx.x * EMB_THREADS + threadIdx.x;
  float s = 0.0f;
#pragma unroll
  for (int sh = 0; sh < EMB_SHARDS; ++sh) s += b[sh * EMB_D_MODEL + d];
  bias_pre[d] = s * 0.125f;
}

// One block per token; each thread moves 4 x float4.
template <bool PRECOMP>
__global__ __launch_bounds__(EMB_THREADS) void gather_rows_kernel(
    const int* __restrict__ tok, const float* __restrict__ W,
    const float* __restrict__ bias_or_b, float* __restrict__ out) {
  const int t   = blockIdx.x;
  const int tid = threadIdx.x;

#if HAVE_ASYNC_LDS
  __shared__ f32x4 lds_bias[PRECOMP ? V4_ROW : 1];
  if (PRECOMP) {
    // Kick off the L2-resident bias tile into LDS; overlaps with the HBM row
    // fetch below. Each lane later reads back only bytes it wrote itself, so
    // s_wait_asynccnt 0 suffices (no workgroup barrier).
#pragma unroll
    for (int k = 0; k < V4_THR; ++k) {
      const int idx = tid + k * EMB_THREADS;
      __builtin_amdgcn_global_load_async_to_lds_b128(
          (gbl_i32x4*)(bias_or_b + (size_t)idx * 4),
          (lds_i32x4*)&lds_bias[idx], /*imm offset*/ 0, /*cpol*/ 0);
    }
  }
#endif

  const int row = tok[t];  // uniform -> scalar load
  const f32x4* __restrict__ src = (const f32x4*)(W + (size_t)row * EMB_D_MODEL);
  f32x4* __restrict__ dst       = (f32x4*)(out + (size_t)t * EMB_D_MODEL);

  // Streamed exactly once -> non-temporal, keep L2 clean.
  f32x4 w[V4_THR];
#pragma unroll
  for (int k = 0; k < V4_THR; ++k)
    w[k] = __builtin_nontemporal_load(src + tid + k * EMB_THREADS);

  f32x4 bb[V4_THR];
  if (PRECOMP) {
#if HAVE_ASYNC_LDS
#if __has_builtin(__builtin_amdgcn_s_wait_asynccnt)
    __builtin_amdgcn_s_wait_asynccnt(0);
#else
    asm volatile("s_wait_asynccnt 0" ::: "memory");
#endif
#pragma unroll
    for (int k = 0; k < V4_THR; ++k) bb[k] = lds_bias[tid + k * EMB_THREADS];
#else
    const f32x4* __restrict__ bsrc = (const f32x4*)bias_or_b;
#pragma unroll
    for (int k = 0; k < V4_THR; ++k) bb[k] = bsrc[tid + k * EMB_THREADS];
#endif
  } else {
    // Fallback (no workspace): sum the 8 shard biases inline (L2-resident).
#pragma unroll
    for (int k = 0; k < V4_THR; ++k) {
      const int idx = tid + k * EMB_THREADS;
      f32x4 s = {};
#pragma unroll
      for (int sh = 0; sh < EMB_SHARDS; ++sh)
        s += ((const f32x4*)(bias_or_b + (size_t)sh * EMB_D_MODEL))[idx];
      bb[k] = s * 0.125f;
    }
  }

#pragma unroll
  for (int k = 0; k < V4_THR; ++k) {
    f32x4 r = w[k] * 0.125f + bb[k];
    __builtin_nontemporal_store(r, dst + tid + k * EMB_THREADS);
  }
}

extern "C" void kernel_launch(void* const* d_in, const int* in_sizes, int n_in,
                              void* d_out, int out_size, void* d_ws,
                              size_t ws_size, hipStream_t stream) {
  (void)n_in; (void)out_size;
  const int*   tok = (const int*)d_in[0];   // x: [B*S] int32
  const float* W   = (const float*)d_in[1]; // [8, 6300, 4096] fp32 (contig == W_full)
  const float* b   = (const float*)d_in[2]; // [8, 4096] fp32
  float*       out = (float*)d_out;         // [B*S, 4096] fp32
  const int ntok = in_sizes[0];             // 16384

  if (ws_size >= (size_t)EMB_D_MODEL * sizeof(float)) {
    float* bias_pre = (float*)d_ws;
    bias_sum_kernel<<<EMB_D_MODEL / EMB_THREADS, EMB_THREADS, 0, stream>>>(b, bias_pre);
    gather_rows_kernel<true><<<ntok, EMB_THREADS, 0, stream>>>(tok, W, bias_pre, out);
  } else {
    gather_rows_kernel<false><<<ntok, EMB_THREADS, 0, stream>>>(tok, W, b, out);
  }
}